// EnhancedOrthogonal_87729001988344
// MI455X (gfx1250) — compile-verified
//
#include <hip/hip_runtime.h>

// EnhancedOrthogonal on MI455X (gfx1250, wave32, WMMA).
// Q lives in d_out ([M,R,2] interleaved f32). ws holds Gre/Gim (R*R each) + norms (R).

typedef __attribute__((ext_vector_type(2))) float v2f;
typedef __attribute__((ext_vector_type(8))) float v8f;

static constexpr int M = 16384;
static constexpr int R = 1024;
static constexpr float EPS = 1e-8f;

__device__ __forceinline__ v8f wmma_f32(v2f a, v2f b, v8f c) {
    // V_WMMA_F32_16X16X4_F32 : D(16x16 f32) = A(16x4 f32) * B(4x16 f32) + C
    return __builtin_amdgcn_wmma_f32_16x16x4_f32(false, a, false, b, (short)0, c,
                                                 false, false);
}

// ---------------------------------------------------------------- utilities
__global__ void zero_kernel(float* __restrict__ p, int n) {
    int i = blockIdx.x * blockDim.x + threadIdx.x;
    if (i < n) p[i] = 0.0f;
}

// Partial column sum-of-squares: grid (R/256, ROW_CHUNKS), block 256.
__global__ __launch_bounds__(256) void colnorm_kernel(const float* __restrict__ src,
                                                      float* __restrict__ norms,
                                                      int rowsPer) {
    int j  = blockIdx.x * 256 + threadIdx.x;
    int r0 = blockIdx.y * rowsPer;
    float s = 0.0f;
    for (int r = 0; r < rowsPer; ++r) {
        const float2 v = *(const float2*)(src + ((size_t)(r0 + r) * R + j) * 2);
        s += v.x * v.x + v.y * v.y;
    }
    atomicAdd(&norms[j], s);
}

// dst = src * rsqrt(norms[col] + eps), elementwise over [M,R] complex.
__global__ __launch_bounds__(256) void scale_kernel(const float* __restrict__ src,
                                                    float* __restrict__ dst,
                                                    const float* __restrict__ norms) {
    const size_t total = (size_t)M * R;
    for (size_t i = (size_t)blockIdx.x * blockDim.x + threadIdx.x; i < total;
         i += (size_t)gridDim.x * blockDim.x) {
        int j = (int)(i & (R - 1));
        float s = rsqrtf(norms[j] + EPS);
        float2 v = *(const float2*)(src + 2 * i);
        v.x *= s;
        v.y *= s;
        *(float2*)(dst + 2 * i) = v;
    }
}

// ---------------------------------------------------------------- gram: H = 0.5*(Q^H Q), diag zeroed
// Block = 128 thr (4 waves) -> 32x32 complex tile, K = M staged through LDS in 32-chunks.
// Hermitian: only bx >= by blocks run; mirror tile written as conj.
__global__ __launch_bounds__(128) void gram_kernel(const float* __restrict__ q,
                                                   float* __restrict__ gre,
                                                   float* __restrict__ gim) {
    if (blockIdx.x < blockIdx.y) return;  // lower triangle handled by mirror write
    const int i0 = blockIdx.y * 32;       // gram row tile
    const int j0 = blockIdx.x * 32;       // gram col tile

    __shared__ float aRe[32][33], aIm[32][33];  // [k][col rel i0], padded
    __shared__ float bRe[32][33], bIm[32][33];  // [k][col rel j0]

    const int tid  = threadIdx.x;  // 0..127
    const int wave = tid >> 5;     // 0..3
    const int lane = tid & 31;
    const int half = lane >> 4;
    const int l16  = lane & 15;
    const int wi   = (wave >> 1) * 16;  // row sub-tile
    const int wj   = (wave & 1) * 16;   // col sub-tile

    v8f accRe = {}, accImP = {}, accImN = {};

    for (int kb = 0; kb < M; kb += 32) {
        // cooperative load: 32 k-rows x 32 cols, complex, for both operand tiles
        for (int e = 0; e < 8; ++e) {
            int idx = tid + e * 128;  // 0..1023
            int kr  = idx >> 5;
            int c   = idx & 31;
            const float2 va = *(const float2*)(q + ((size_t)(kb + kr) * R + (i0 + c)) * 2);
            const float2 vb = *(const float2*)(q + ((size_t)(kb + kr) * R + (j0 + c)) * 2);
            aRe[kr][c] = va.x; aIm[kr][c] = va.y;
            bRe[kr][c] = vb.x; bIm[kr][c] = vb.y;
        }
        __syncthreads();
        for (int k0 = 0; k0 < 32; k0 += 4) {
            const int kk = k0 + half * 2;  // lanes 0-15 -> k,k+1 ; lanes 16-31 -> k+2,k+3
            v2f ar, ai, br, bi;
            ar.x = aRe[kk][wi + l16]; ar.y = aRe[kk + 1][wi + l16];
            ai.x = aIm[kk][wi + l16]; ai.y = aIm[kk + 1][wi + l16];
            br.x = bRe[kk][wj + l16]; br.y = bRe[kk + 1][wj + l16];
            bi.x = bIm[kk][wj + l16]; bi.y = bIm[kk + 1][wj + l16];
            accRe  = wmma_f32(ar, br, accRe);   // Re += ArBr
            accRe  = wmma_f32(ai, bi, accRe);   // Re += AiBi
            accImP = wmma_f32(ar, bi, accImP);  // Im += ArBi
            accImN = wmma_f32(ai, br, accImN);  // Im -= AiBr (subtracted below)
        }
        __syncthreads();
    }

    // writeout: H = 0.5 * gram_off; C layout: row = v + 8*half, col = l16
    for (int v = 0; v < 8; ++v) {
        const int gi = i0 + wi + v + 8 * half;
        const int gj = j0 + wj + l16;
        float re = 0.5f * accRe[v];
        float im = 0.5f * (accImP[v] - accImN[v]);
        if (gi == gj) { re = 0.0f; im = 0.0f; }
        gre[(size_t)gi * R + gj] = re;
        gim[(size_t)gi * R + gj] = im;
        if (i0 != j0) {  // Hermitian mirror
            gre[(size_t)gj * R + gi] = re;
            gim[(size_t)gj * R + gi] = -im;
        }
    }
}

// ---------------------------------------------------------------- update: Q -= Q @ H  (row-local, in place)
// Block = 256 thr (8 waves) owns a 16-row band; whole band staged in 128KB LDS,
// then overwritten in global. Also accumulates new column sum-squares.
__global__ __launch_bounds__(256) void update_kernel(float* __restrict__ q,
                                                     const float* __restrict__ gre,
                                                     const float* __restrict__ gim,
                                                     float* __restrict__ norms) {
    extern __shared__ float lds[];          // bandRe[1024][16] ++ bandIm[1024][16]
    float* bandRe = lds;                    // k-major: [k][m], conflict-free frag reads
    float* bandIm = lds + (size_t)R * 16;

    const int m0   = blockIdx.x * 16;
    const int tid  = threadIdx.x;
    const int wave = tid >> 5;   // 0..7
    const int lane = tid & 31;
    const int half = lane >> 4;
    const int l16  = lane & 15;

    // stage band: 16 rows x 1024 cols complex (coalesced b64 reads)
    for (int e = 0; e < 64; ++e) {
        int idx = tid + e * 256;   // 0..16383
        int mr  = idx >> 10;       // band row
        int k   = idx & 1023;      // Q column
        const float2 v = *(const float2*)(q + ((size_t)(m0 + mr) * R + k) * 2);
        bandRe[k * 16 + mr] = v.x;
        bandIm[k * 16 + mr] = v.y;
    }
    __syncthreads();

    for (int pass = 0; pass < 8; ++pass) {
        const int j0 = pass * 128 + wave * 16;  // this wave's output column tile
        v8f accA = {}, accB = {}, accC = {};    // QreHre, QimHim, Qre*Him + Qim*Hre
        for (int k0 = 0; k0 < R; k0 += 4) {
            const int kk = k0 + half * 2;
            v2f ar, ai, br, bi;
            ar.x = bandRe[kk * 16 + l16];       ar.y = bandRe[(kk + 1) * 16 + l16];
            ai.x = bandIm[kk * 16 + l16];       ai.y = bandIm[(kk + 1) * 16 + l16];
            const float* hre = gre + (size_t)kk * R + (j0 + l16);  // H is L2-resident (8MB)
            const float* him = gim + (size_t)kk * R + (j0 + l16);
            br.x = hre[0]; br.y = hre[R];
            bi.x = him[0]; bi.y = him[R];
            accA = wmma_f32(ar, br, accA);
            accB = wmma_f32(ai, bi, accB);
            accC = wmma_f32(ar, bi, accC);
            accC = wmma_f32(ai, br, accC);
        }
        // Qnew = Qold - P ; P.re = accA - accB, P.im = accC
        float colsum = 0.0f;
        const int j = j0 + l16;
        for (int v = 0; v < 8; ++v) {
            const int mr = v + 8 * half;
            float qre = bandRe[j * 16 + mr] - (accA[v] - accB[v]);
            float qim = bandIm[j * 16 + mr] - accC[v];
            float2 o; o.x = qre; o.y = qim;
            *(float2*)(q + ((size_t)(m0 + mr) * R + j) * 2) = o;
            colsum += qre * qre + qim * qim;
        }
        atomicAdd(&norms[j], colsum);  // 2 lanes/column/block
    }
}

// ---------------------------------------------------------------- launch
extern "C" void kernel_launch(void* const* d_in, const int* in_sizes, int n_in,
                              void* d_out, int out_size, void* d_ws, size_t ws_size,
                              hipStream_t stream) {
    (void)in_sizes; (void)n_in; (void)out_size; (void)ws_size;
    const float* x = (const float*)d_in[0];   // [M,R,2] f32
    float* q = (float*)d_out;                 // Q lives here, [M,R,2]
    float* gre   = (float*)d_ws;              // R*R
    float* gim   = gre + (size_t)R * R;       // R*R
    float* norms = gim + (size_t)R * R;       // R       (total ws ~ 8.4 MB)

    const int ROWS_PER = 128;  // colnorm row chunk

    // initial column normalization: Q = x * rsqrt(sumsq + eps)
    zero_kernel<<<(R + 255) / 256, 256, 0, stream>>>(norms, R);
    colnorm_kernel<<<dim3(R / 256, M / ROWS_PER), 256, 0, stream>>>(x, norms, ROWS_PER);
    scale_kernel<<<4096, 256, 0, stream>>>(x, q, norms);

    for (int it = 0; it < 2; ++it) {
        gram_kernel<<<dim3(R / 32, R / 32), 128, 0, stream>>>(q, gre, gim);
        zero_kernel<<<(R + 255) / 256, 256, 0, stream>>>(norms, R);
        update_kernel<<<M / 16, 256, 2u * R * 16 * sizeof(float), stream>>>(q, gre, gim, norms);
        scale_kernel<<<4096, 256, 0, stream>>>(q, q, norms);
    }
}